// NeighborCooccurrenceEncoder_74543452389427
// MI455X (gfx1250) — compile-verified
//
#include <hip/hip_runtime.h>

#define L 256
#define F 64
#define BM_WORDS 512   // 16384-bit bitmap; ids are in [0, 10000)

typedef __attribute__((ext_vector_type(4))) float v4f;

__global__ __launch_bounds__(256) void cooccur_feat_kernel(
    const int* __restrict__ src_ids, const int* __restrict__ dst_ids,
    const float* __restrict__ scale, float* __restrict__ out, int B) {
  __shared__ unsigned bm_s[BM_WORDS];  // bitmap of src ids in this row
  __shared__ unsigned bm_d[BM_WORDS];  // bitmap of dst ids in this row
  __shared__ int ids_s[L];
  __shared__ int ids_d[L];
  __shared__ float m_s[L];
  __shared__ float m_d[L];

  const int t = threadIdx.x;
  const int r = blockIdx.x;

  // Zero the bitmaps (1024 dwords / 256 threads).
  bm_s[t] = 0u; bm_s[t + 256] = 0u;
  bm_d[t] = 0u; bm_d[t + 256] = 0u;

  // Stage both id rows into LDS with CDNA5 async global->LDS loads.
  // Each lane fetches one 4-byte id per row; tracked by ASYNCcnt.
  {
    const int* gs = src_ids + (size_t)r * L;
    const int* gd = dst_ids + (size_t)r * L;
    unsigned voff = (unsigned)(t * 4);
    // Flat shared-pointer low 32 bits == wave-relative LDS byte address
    // (aperture mapping: the low 32 bits give the LDS address).
    unsigned lds_s_addr = (unsigned)(unsigned long long)&ids_s[t];
    unsigned lds_d_addr = (unsigned)(unsigned long long)&ids_d[t];
    asm volatile("global_load_async_to_lds_b32 %0, %1, %2"
                 :: "v"(lds_s_addr), "v"(voff), "s"(gs) : "memory");
    asm volatile("global_load_async_to_lds_b32 %0, %1, %2"
                 :: "v"(lds_d_addr), "v"(voff), "s"(gd) : "memory");
    asm volatile("s_wait_asynccnt 0" ::: "memory");
  }
  __syncthreads();

  // Set membership bits (ds_or_b32).
  const int a = ids_s[t];
  const int b = ids_d[t];
  atomicOr(&bm_s[(a >> 5) & (BM_WORDS - 1)], 1u << (a & 31));
  atomicOr(&bm_d[(b >> 5) & (BM_WORDS - 1)], 1u << (b & 31));
  __syncthreads();

  // O(1) membership probe per element (replaces the O(L) compare loop).
  const float s = scale[0];
  m_s[t] = ((bm_d[(a >> 5) & (BM_WORDS - 1)] >> (a & 31)) & 1u) ? s : 0.0f;
  m_d[t] = ((bm_s[(b >> 5) & (BM_WORDS - 1)] >> (b & 31)) & 1u) ? s : 0.0f;
  __syncthreads();

  // Streaming output: [B, L, 64] f32, src tensor then dst tensor.
  // Per row-block: 4096 float4s per direction, fully coalesced.
  // Non-temporal: 512 MB total can never live in 192 MB L2.
  v4f* out_s = (v4f*)(out + (size_t)r * (L * F));
  v4f* out_d = (v4f*)(out + (size_t)B * (L * F) + (size_t)r * (L * F));
#pragma unroll
  for (int it = 0; it < 16; ++it) {
    int f = it * 256 + t;        // float4 index within this row's block
    int j = f >> 4;              // which of the 256 elements
    bool ch0 = (f & 15) == 0;    // float4 containing channel 0
    v4f v = {ch0 ? m_s[j] : 0.0f, 0.0f, 0.0f, 0.0f};
    v4f w = {ch0 ? m_d[j] : 0.0f, 0.0f, 0.0f, 0.0f};
    __builtin_nontemporal_store(v, &out_s[f]);
    __builtin_nontemporal_store(w, &out_d[f]);
  }
}

extern "C" void kernel_launch(void* const* d_in, const int* in_sizes, int n_in,
                              void* d_out, int out_size, void* d_ws, size_t ws_size,
                              hipStream_t stream) {
  const int* src_ids = (const int*)d_in[0];
  const int* dst_ids = (const int*)d_in[1];
  const float* scale = (const float*)d_in[2];
  float* out = (float*)d_out;
  const int B = in_sizes[0] / L;  // 4096
  hipLaunchKernelGGL(cooccur_feat_kernel, dim3(B), dim3(256), 0, stream,
                     src_ids, dst_ids, scale, out, B);
}